// GMM_87170656240130
// MI455X (gfx1250) — compile-verified
//
#include <hip/hip_runtime.h>
#include <math.h>

typedef __attribute__((ext_vector_type(2))) float v2f;
typedef __attribute__((ext_vector_type(8))) float v8f;

#define LOG_2PI 1.8378770664093453f
#define LOG2E   1.4426950408889634f
#define LN2     0.6931471805599453f

// ---------------------------------------------------------------------------
// Setup: one wave (K=32 lanes). Coefficients pre-scaled by log2(e) so WMMA
// produces log2-probs, pre-shifted by M = max_k c_k (exp2 args are not bounded above by 0:
// the x terms can exceed M - c_k; no per-row max pass is done).
//   G[0*32+k] = -0.5*iv0*LOG2E    (coeff of x0^2)
//   G[1*32+k] = -0.5*iv1*LOG2E    (coeff of x1^2)
//   G[2*32+k] =  m0*iv0*LOG2E     (coeff of x0)
//   G[3*32+k] =  m1*iv1*LOG2E     (coeff of x1)
//   G[4*32+k] = (c_k - M)*LOG2E   (accumulator init, row k)
//   G[160]    =  M
// ---------------------------------------------------------------------------
__global__ __launch_bounds__(32) void gmm_setup_kernel(
    const float* __restrict__ w, const float* __restrict__ means,
    const float* __restrict__ logvar, float* __restrict__ G) {
  const int k = threadIdx.x;  // 0..31 == K
  float wk = w[k];

  float mx = wk;
#pragma unroll
  for (int m = 16; m >= 1; m >>= 1) mx = fmaxf(mx, __shfl_xor(mx, m, 32));
  float s = __expf(wk - mx);
#pragma unroll
  for (int m = 16; m >= 1; m >>= 1) s += __shfl_xor(s, m, 32);
  const float logw = wk - mx - __logf(s);

  const float lv0 = logvar[2 * k], lv1 = logvar[2 * k + 1];
  const float iv0 = __expf(-lv0), iv1 = __expf(-lv1);
  const float m0 = means[2 * k], m1 = means[2 * k + 1];

  const float ck = logw - 0.5f * (lv0 + lv1 + 2.0f * LOG_2PI)
                        - 0.5f * (m0 * m0 * iv0 + m1 * m1 * iv1);
  float M = ck;
#pragma unroll
  for (int m = 16; m >= 1; m >>= 1) M = fmaxf(M, __shfl_xor(M, m, 32));

  G[0 * 32 + k] = -0.5f * iv0 * LOG2E;
  G[1 * 32 + k] = -0.5f * iv1 * LOG2E;
  G[2 * 32 + k] = m0 * iv0 * LOG2E;
  G[3 * 32 + k] = m1 * iv1 * LOG2E;
  G[4 * 32 + k] = (ck - M) * LOG2E;
  if (k == 0) G[160] = M;
}

// ---------------------------------------------------------------------------
// Main kernel, transposed GEMM: D[k, n] = 32 components (rows, two WMMAs of
// 16) x 16 points (columns) per tile.
//   A (16x4, loop-invariant): row = component, K = feature; lanes 0-15 hold
//       K=0,1 coeffs, lanes 16-31 hold K=2,3 coeffs for component col=lane&15.
//   B (4x16, per tile): column = point (lane&15); lanes 0-15 supply rows
//       K=0,1 = {x0^2, x1^2}, lanes 16-31 rows K=2,3 = {x0, x1}.
//   C: per-ROW constants (c_k - M)*LOG2E; VGPR j = row j (lo) / j+8 (hi).
// Reduction over 32 components = in-lane sum of the 16 D registers (lo half
// holds comps 0-7 & 16-23 of its point, hi half comps 8-15 & 24-31) plus a
// single shfl_xor(16). Lanes 0-15 store 16 contiguous results.
// ---------------------------------------------------------------------------
__global__ __launch_bounds__(256) void gmm_wmma_kernel(
    const float* __restrict__ Xf, const float* __restrict__ G,
    float* __restrict__ out, int npoints, int ntiles) {
  const v2f* __restrict__ X = (const v2f*)Xf;
  const int lane = threadIdx.x & 31;
  const int col = lane & 15;
  const bool hi = lane >= 16;

  const int wavesPerBlock = blockDim.x >> 5;
  const int gwave = blockIdx.x * wavesPerBlock + (threadIdx.x >> 5);
  const int nwaves = gridDim.x * wavesPerBlock;

  // Loop-invariant A operands (components 0-15 / 16-31).
  v2f aL, aH;
  aL.x = hi ? G[2 * 32 + col] : G[0 * 32 + col];             // K = hi?2:0
  aL.y = hi ? G[3 * 32 + col] : G[1 * 32 + col];             // K = hi?3:1
  aH.x = hi ? G[2 * 32 + 16 + col] : G[0 * 32 + 16 + col];
  aH.y = hi ? G[3 * 32 + 16 + col] : G[1 * 32 + 16 + col];

  // Loop-invariant per-row accumulator constants.
  v8f cL, cH;
#pragma unroll
  for (int j = 0; j < 8; ++j) {
    cL[j] = hi ? G[4 * 32 + j + 8] : G[4 * 32 + j];
    cH[j] = hi ? G[4 * 32 + 16 + j + 8] : G[4 * 32 + 16 + j];
  }
  const float M = G[160];

  for (int t = gwave; t < ntiles; t += nwaves) {
    int p = t * 16 + col;
    int pc = p < npoints ? p : npoints - 1;  // clamp tail (keeps EXEC full)
    v2f x = X[pc];

    // branchless prefetch of the next grid-stride tile (global_prefetch_b8)
    int pf = p + nwaves * 16;
    pf = pf < npoints ? pf : npoints - 1;
    __builtin_prefetch(&X[pf], 0, 1);

    // B: feature rows for the 16 points of this tile.
    v2f b;
    b.x = hi ? x.x : x.x * x.x;  // K = hi?2:0
    b.y = hi ? x.y : x.y * x.y;  // K = hi?3:1

    v8f d1 = __builtin_amdgcn_wmma_f32_16x16x4_f32(
        false, aL, false, b, (short)0, cL, false, false);   // comps 0-15
    v8f d2 = __builtin_amdgcn_wmma_f32_16x16x4_f32(
        false, aH, false, b, (short)0, cH, false, false);   // comps 16-31

    // In-lane sum of 16 exp2's (bare v_exp_f32), balanced tree.
    float s0 = __builtin_amdgcn_exp2f(d1[0]) + __builtin_amdgcn_exp2f(d2[0]);
    float s1 = __builtin_amdgcn_exp2f(d1[1]) + __builtin_amdgcn_exp2f(d2[1]);
    float s2 = __builtin_amdgcn_exp2f(d1[2]) + __builtin_amdgcn_exp2f(d2[2]);
    float s3 = __builtin_amdgcn_exp2f(d1[3]) + __builtin_amdgcn_exp2f(d2[3]);
    float s4 = __builtin_amdgcn_exp2f(d1[4]) + __builtin_amdgcn_exp2f(d2[4]);
    float s5 = __builtin_amdgcn_exp2f(d1[5]) + __builtin_amdgcn_exp2f(d2[5]);
    float s6 = __builtin_amdgcn_exp2f(d1[6]) + __builtin_amdgcn_exp2f(d2[6]);
    float s7 = __builtin_amdgcn_exp2f(d1[7]) + __builtin_amdgcn_exp2f(d2[7]);
    float t0 = s0 + s1, t1 = s2 + s3, t2 = s4 + s5, t3 = s6 + s7;
    float part = (t0 + t1) + (t2 + t3);

    // Combine the two half-waves: full 32-component sum for point `col`.
    float ssum = part + __shfl_xor(part, 16, 32);

    float res = fmaf(__builtin_amdgcn_logf(ssum), LN2, M);

    int idx = t * 16 + col;
    if (!hi && idx < npoints) out[idx] = res;  // 16 contiguous floats/wave
  }
}

extern "C" void kernel_launch(void* const* d_in, const int* in_sizes, int n_in,
                              void* d_out, int out_size, void* d_ws, size_t ws_size,
                              hipStream_t stream) {
  const float* X = (const float*)d_in[0];       // [N,2]
  const float* w = (const float*)d_in[1];       // [32]
  const float* means = (const float*)d_in[2];   // [32,2]
  const float* logvar = (const float*)d_in[3];  // [32,2]
  float* out = (float*)d_out;                   // [N]
  float* G = (float*)d_ws;                      // 5*32 floats + M

  const int npoints = in_sizes[0] / 2;
  const int ntiles = (npoints + 15) / 16;

  gmm_setup_kernel<<<1, 32, 0, stream>>>(w, means, logvar, G);

  int blocks = (ntiles + 7) / 8;    // 8 waves/block, 1 tile per wave per pass
  if (blocks > 8192) blocks = 8192; // persistent waves, grid-stride
  if (blocks < 1) blocks = 1;
  gmm_wmma_kernel<<<blocks, 256, 0, stream>>>(X, G, out, npoints, ntiles);
}